// NeuralODEDiffrax_86406152061714
// MI455X (gfx1250) — compile-verified
//
#include <hip/hip_runtime.h>

typedef float v2f __attribute__((ext_vector_type(2)));
typedef float v8f __attribute__((ext_vector_type(8)));

#define WAVES_PER_BLOCK 8
#define THREADS_PER_BLOCK (WAVES_PER_BLOCK * 32)
#define N_BATCH 65536
#define N_SAVE 100
#define H_STRIDE 36  // row stride (floats): 8*36 % 64 == 32 -> b64 store & load conflict-free

// ---- Dopri5 tableau ----
#define A21 0.2f
#define A31 (3.0f/40.0f)
#define A32 (9.0f/40.0f)
#define A41 (44.0f/45.0f)
#define A42 (-56.0f/15.0f)
#define A43 (32.0f/9.0f)
#define A51 (19372.0f/6561.0f)
#define A52 (-25360.0f/2187.0f)
#define A53 (64448.0f/6561.0f)
#define A54 (-212.0f/729.0f)
#define A61 (9017.0f/3168.0f)
#define A62 (-355.0f/33.0f)
#define A63 (46732.0f/5247.0f)
#define A64 (49.0f/176.0f)
#define A65 (-5103.0f/18656.0f)
#define B1  (35.0f/384.0f)
#define B3  (500.0f/1113.0f)
#define B4  (125.0f/192.0f)
#define B5  (-2187.0f/6784.0f)
#define B6  (11.0f/84.0f)
#define E1  (71.0f/57600.0f)
#define E3  (-71.0f/16695.0f)
#define E4  (71.0f/1920.0f)
#define E5  (-17253.0f/339200.0f)
#define E6  (22.0f/525.0f)
#define E7  (-1.0f/40.0f)
#define RTOL 1e-6f
#define ATOL 1e-6f

#define LOG2E 1.4426950408889634f
#define LN2   0.6931471805599453f

__device__ __forceinline__ float raw_exp2(float x) {
#if __has_builtin(__builtin_amdgcn_exp2f)
    return __builtin_amdgcn_exp2f(x);      // v_exp_f32, no guard
#else
    return __expf(x * LN2);
#endif
}

__device__ __forceinline__ float raw_log2(float x) {
#if __has_builtin(__builtin_amdgcn_logf)
    return __builtin_amdgcn_logf(x);       // v_log_f32, no denorm guard
#else
    return __logf(x) * LOG2E;
#endif
}

__device__ __forceinline__ float raw_rcp(float x) {
#if __has_builtin(__builtin_amdgcn_rcpf)
    return __builtin_amdgcn_rcpf(x);
#else
    return 1.0f / x;
#endif
}

__device__ __forceinline__ float raw_sqrt(float x) {
#if __has_builtin(__builtin_amdgcn_sqrtf)
    return __builtin_amdgcn_sqrtf(x);
#else
    return sqrtf(x);
#endif
}

__device__ __forceinline__ float fast_softplus(float x) {
    // max(x,0) + ln2 * log2(1 + 2^(-|x|*log2e)); log2 input in [1,2] -> guard-free
    float e = raw_exp2(-fabsf(x) * LOG2E);
    return fmaxf(x, 0.0f) + LN2 * raw_log2(1.0f + e);
}

__device__ __forceinline__ float fast_tanh(float x) {
#if __has_builtin(__builtin_amdgcn_tanhf)
    return __builtin_amdgcn_tanhf(x);      // native v_tanh_f32 on gfx1250
#else
    float e = raw_exp2(-2.0f * LOG2E * fabsf(x));
    float t = (1.0f - e) * raw_rcp(1.0f + e);
    return copysignf(t, x);
#endif
}

__global__ __launch_bounds__(THREADS_PER_BLOCK)
void node_dopri5_kernel(const float* __restrict__ y0g,
                        const float* __restrict__ t1p,
                        const float* __restrict__ w1,  // (32,4) row-major
                        const float* __restrict__ b1,  // (32,)
                        const float* __restrict__ w2,  // (4,32) row-major
                        const float* __restrict__ b2,  // (4,)
                        float* __restrict__ partials)
{
    __shared__ float ldsHbuf[WAVES_PER_BLOCK][16 * H_STRIDE];
    __shared__ float ldsZbuf[WAVES_PER_BLOCK][16 * 4];

    const int lane  = threadIdx.x & 31;
    const int wib   = threadIdx.x >> 5;
    const int wave  = blockIdx.x * WAVES_PER_BLOCK + wib;
    float* ldsH = ldsHbuf[wib];
    float* ldsZ = ldsZbuf[wib];

    const int mrow  = lane & 15;      // state row within wave tile / N index
    const int khalf = lane >> 4;      // 0: comps {0,1}; 1: comps {2,3}

    // ---- loop-invariant B-format weights ----
    // Hidden-unit permutation: layer-1 tile t, column n computes hidden unit j = 2n + t.
    // (MLP invariant under consistent permutation of hidden units; makes the H
    //  transpose store a single b64 per lane per row.)
    v2f Bw1[2];
#pragma unroll
    for (int t = 0; t < 2; ++t) {
        int j  = 2 * mrow + t;
        int k0 = khalf * 2;
        v2f v; v.x = w1[j * 4 + k0]; v.y = w1[j * 4 + k0 + 1];
        Bw1[t] = v;
    }
    const float bias1a = b1[2 * mrow];
    const float bias1b = b1[2 * mrow + 1];

    // Layer 2: B = W2^T (32x4) as eight 4x16 chunks, natural hidden order (cols >=4 zero).
    v2f Bw2[8];
#pragma unroll
    for (int c = 0; c < 8; ++c) {
        int kk0 = 4 * c + khalf * 2;
        v2f v; v.x = 0.0f; v.y = 0.0f;
        if (mrow < 4) { v.x = w2[mrow * 32 + kk0]; v.y = w2[mrow * 32 + kk0 + 1]; }
        Bw2[c] = v;
    }
    const float bias2 = (mrow < 4) ? b2[mrow] : 0.0f;

    // ---- hoisted per-lane LDS base pointers (imm offsets inside the loop) ----
    float* pStoreH = ldsH + khalf * 8 * H_STRIDE + 2 * mrow;     // + r*H_STRIDE
    float* pLoadH  = ldsH + mrow * H_STRIDE + khalf * 2;          // + 4*c
    float* pStoreZ = ldsZ + khalf * 8 * 4 + mrow;                 // + r*4   (mrow<4 only)
    float* pLoadZ  = ldsZ + mrow * 4 + khalf * 2;

    // ---- initial state in A-format ----
    const int state = wave * 16 + mrow;
    v2f y = *(const v2f*)(y0g + state * 4 + khalf * 2);
    const float t1 = *t1p;

    // ---- MLP vector field: 2 WMMA (layer1) + LDS transpose + 8 WMMA (layer2) ----
    auto vf = [&](v2f yin) -> v2f {
        v8f h0 = {}; v8f h1 = {};
        h0 = __builtin_amdgcn_wmma_f32_16x16x4_f32(false, yin, false, Bw1[0],
                                                   (short)0, h0, false, false);
        h1 = __builtin_amdgcn_wmma_f32_16x16x4_f32(false, yin, false, Bw1[1],
                                                   (short)0, h1, false, false);
        // bias + softplus; adjacent hidden-unit pair -> one b64 store per row
#pragma unroll
        for (int r = 0; r < 8; ++r) {
            v2f sp;
            sp.x = fast_softplus(h0[r] + bias1a);
            sp.y = fast_softplus(h1[r] + bias1b);
            *(v2f*)(pStoreH + r * H_STRIDE) = sp;
        }
        // layer 2: K accumulation over 8 chunks, A re-read in A-format (b64 loads)
        v8f z = {};
#pragma unroll
        for (int c = 0; c < 8; ++c) {
            v2f a = *(const v2f*)(pLoadH + 4 * c);
            z = __builtin_amdgcn_wmma_f32_16x16x4_f32(false, a, false, Bw2[c],
                                                      (short)0, z, false, false);
        }
        // bias + tanh on the 4 live output columns, route back to A-format via LDS
        if (mrow < 4) {
#pragma unroll
            for (int r = 0; r < 8; ++r) {
                pStoreZ[r * 4] = fast_tanh(z[r] + bias2);
            }
        }
        return *(const v2f*)pLoadZ;
    };

    float acc = y.x + y.y;      // save at t = 0
    v2f k1 = vf(y);             // FSAL seed
    float t  = 0.0f;
    float dt = 0.01f * t1;
    if (!(dt > 0.0f)) dt = 0.01f;

    for (int s = 1; s < N_SAVE; ++s) {
        const float tend = t1 * ((float)s / (float)(N_SAVE - 1));
        for (int iter = 0; iter < 100000; ++iter) {
            float rem = tend - t;
            if (rem <= 1e-9f) break;
            float h = fminf(dt, rem);

            v2f k2 = vf(y + (h * A21) * k1);
            v2f k3 = vf(y + h * (A31 * k1 + A32 * k2));
            v2f k4 = vf(y + h * (A41 * k1 + A42 * k2 + A43 * k3));
            v2f k5 = vf(y + h * (A51 * k1 + A52 * k2 + A53 * k3 + A54 * k4));
            v2f k6 = vf(y + h * (A61 * k1 + A62 * k2 + A63 * k3 + A64 * k4 + A65 * k5));
            v2f ynew = y + h * (B1 * k1 + B3 * k3 + B4 * k4 + B5 * k5 + B6 * k6);
            v2f k7 = vf(ynew);
            v2f err = h * (E1 * k1 + E3 * k3 + E4 * k4 + E5 * k5 + E6 * k6 + E7 * k7);

            // per-state RMS of err/scale over 4 components (split across lane pairs)
            float s0 = ATOL + RTOL * fmaxf(fabsf(y.x), fabsf(ynew.x));
            float s1 = ATOL + RTOL * fmaxf(fabsf(y.y), fabsf(ynew.y));
            float r0 = err.x * raw_rcp(s0), r1 = err.y * raw_rcp(s1);
            float sum2 = r0 * r0 + r1 * r1;
            sum2 += __shfl_xor(sum2, 16);
            float norm = raw_sqrt(0.25f * sum2);
            // wave-shared step control: most conservative state wins
#pragma unroll
            for (int d = 1; d < 32; d <<= 1) norm = fmaxf(norm, __shfl_xor(norm, d));

            float factor = (norm < 1e-8f)
                               ? 10.0f
                               : fminf(10.0f, fmaxf(0.2f,
                                     0.9f * raw_exp2(-0.2f * raw_log2(norm))));
            bool accept  = (norm <= 1.0f);
            bool clamped = (dt >= rem);
            if (accept) {
                t = clamped ? tend : (t + h);
                y = ynew;
                k1 = k7;   // FSAL
            }
            float dtn = h * factor;
            if (accept && clamped) dtn = fmaxf(dtn, dt);  // keep dt across boundary clamp
            dt = fmaxf(dtn, 1e-7f);
        }
        acc += y.x + y.y;   // save at tend
    }

    // wave-level sum of this wave's 16 states x 4 comps x 100 saves
#pragma unroll
    for (int d = 1; d < 32; d <<= 1) acc += __shfl_xor(acc, d);
    if (lane == 0) partials[wave] = acc;
}

__global__ __launch_bounds__(256)
void reduce_kernel(const float* __restrict__ partials, float* __restrict__ out, int n)
{
    __shared__ float buf[256];
    int tid = threadIdx.x;
    float s = 0.0f;
    for (int i = tid; i < n; i += 256) s += partials[i];  // fixed order -> deterministic
    buf[tid] = s;
    __syncthreads();
    for (int w = 128; w > 0; w >>= 1) {
        if (tid < w) buf[tid] += buf[tid + w];
        __syncthreads();
    }
    if (tid == 0) out[0] = buf[0];
}

extern "C" void kernel_launch(void* const* d_in, const int* in_sizes, int n_in,
                              void* d_out, int out_size, void* d_ws, size_t ws_size,
                              hipStream_t stream) {
    const float* y0 = (const float*)d_in[0];
    const float* t1 = (const float*)d_in[1];
    const float* w1 = (const float*)d_in[2];
    const float* b1 = (const float*)d_in[3];
    const float* w2 = (const float*)d_in[4];
    const float* b2 = (const float*)d_in[5];

    float* partials = (float*)d_ws;              // 4096 floats
    const int nWaves = N_BATCH / 16;             // 4096
    const int blocks = nWaves / WAVES_PER_BLOCK; // 512

    node_dopri5_kernel<<<blocks, THREADS_PER_BLOCK, 0, stream>>>(
        y0, t1, w1, b1, w2, b2, partials);
    reduce_kernel<<<1, 256, 0, stream>>>(partials, (float*)d_out, nWaves);
}